// LightGCN_78176994722033
// MI455X (gfx1250) — compile-verified
//
#include <hip/hip_runtime.h>
#include <hip/hip_bf16.h>

#define NUM_USERS 100000
#define NUM_ITEMS 100000
#define NUM_NODES (NUM_USERS + NUM_ITEMS)
#define EMB_DIM   64
#define NUM_EDGES 4000000
#define BATCH     4096

typedef __attribute__((ext_vector_type(2))) float v2f;
typedef __attribute__((ext_vector_type(8))) float v8f;

// ---------------------------------------------------------------------------
// Zero the two SpMM accumulation buffers (float4-vectorized).
// ---------------------------------------------------------------------------
__global__ void lgcn_zero_kernel(float4* __restrict__ p, int n4) {
    int i = blockIdx.x * blockDim.x + threadIdx.x;
    if (i < n4) p[i] = make_float4(0.f, 0.f, 0.f, 0.f);
}

// ---------------------------------------------------------------------------
// Edge-parallel SpMM: one wave (32 lanes) per edge; each lane owns a
// contiguous float2 of the 64-dim row (global_load_b64), then two f32
// hardware atomics into the destination row. Source is split into two
// pointers so the same kernel handles concat(user_emb,item_emb) in layer 1
// and the contiguous ping buffer in layer 2.
// ---------------------------------------------------------------------------
__global__ void lgcn_spmm_kernel(const int*   __restrict__ erow,
                                 const int*   __restrict__ ecol,
                                 const float* __restrict__ eval,
                                 const float* __restrict__ srcA,   // cols [0, NUM_USERS)
                                 const float* __restrict__ srcB,   // cols [NUM_USERS, NUM_NODES)
                                 float*       __restrict__ dst) {
    unsigned tid = blockIdx.x * blockDim.x + threadIdx.x;
    int e = (int)(tid >> 5);
    if (e >= NUM_EDGES) return;
    int d = (int)(tid & 31u) << 1;            // lane's 2-float slice of the row

    int   r = erow[e];
    int   c = ecol[e];
    float v = eval[e];

    const float* src = (c < NUM_USERS)
                         ? (srcA + (size_t)c * EMB_DIM)
                         : (srcB + (size_t)(c - NUM_USERS) * EMB_DIM);
    float2 x = *(const float2*)(src + d);

    float* dp = dst + (size_t)r * EMB_DIM + d;
    unsafeAtomicAdd(dp,     v * x.x);         // global_atomic_add_f32
    unsafeAtomicAdd(dp + 1, v * x.y);
}

// ---------------------------------------------------------------------------
// Diagonal extraction helper for the 16x16 f32 C/D accumulator layout.
// ---------------------------------------------------------------------------
__device__ __forceinline__ float lgcn_vext(v8f v, int i) {
    float r = v[0];
#pragma unroll
    for (int j = 1; j < 8; ++j) r = (i == j) ? v[j] : r;
    return r;
}

// ---------------------------------------------------------------------------
// Scoring: one wave per 16 batch rows. Latent row = (emb0 + h1 + h2)/3,
// assembled on the fly. scores = diag(U @ P^T) computed with 16 chained
// V_WMMA_F32_16X16X4_F32 ops (K = 4 per step, 64 total), two accumulators
// (pos / neg) sharing the A operand.
//
// A-matrix 16x4 f32 layout: lane L holds row M=L&15, K-pair (2*kh, 2*kh+1)
// with kh = L>>4, in VGPR0/VGPR1. B-matrix 4x16 mirrors it (row K striped
// across lanes). C/D: VGPR r = rows r (lanes 0-15) and r+8 (lanes 16-31).
// EXEC is all-1s: grid is an exact multiple (256 waves), no divergence
// before/inside the WMMA loop.
// ---------------------------------------------------------------------------
__global__ void lgcn_score_kernel(const int*   __restrict__ user,
                                  const int*   __restrict__ pos,
                                  const int*   __restrict__ neg,
                                  const float* __restrict__ user_emb,
                                  const float* __restrict__ item_emb,
                                  const float* __restrict__ buf1,
                                  const float* __restrict__ buf2,
                                  float*       __restrict__ out) {
    int wave = (blockIdx.x * blockDim.x + threadIdx.x) >> 5;   // 0..255
    int lane = threadIdx.x & 31;
    int m    = lane & 15;
    int kh   = lane >> 4;
    int b    = wave * 16 + m;

    int uidx = user[b];
    int pidx = pos[b];
    int nidx = neg[b];

    const float* ue = user_emb + (size_t)uidx * EMB_DIM;
    const float* u1 = buf1     + (size_t)uidx * EMB_DIM;
    const float* u2 = buf2     + (size_t)uidx * EMB_DIM;

    const float* pe = item_emb + (size_t)pidx * EMB_DIM;
    const float* p1 = buf1     + (size_t)(NUM_USERS + pidx) * EMB_DIM;
    const float* p2 = buf2     + (size_t)(NUM_USERS + pidx) * EMB_DIM;

    const float* ge = item_emb + (size_t)nidx * EMB_DIM;
    const float* g1 = buf1     + (size_t)(NUM_USERS + nidx) * EMB_DIM;
    const float* g2 = buf2     + (size_t)(NUM_USERS + nidx) * EMB_DIM;

    const float inv3 = 1.0f / 3.0f;
    v8f cp = {};   // pos-score accumulator (16x16 f32 tile)
    v8f cn = {};   // neg-score accumulator

#pragma unroll
    for (int s = 0; s < 16; ++s) {
        int k = s * 4 + kh * 2;

        float2 ae = *(const float2*)(ue + k);
        float2 a1 = *(const float2*)(u1 + k);
        float2 a2 = *(const float2*)(u2 + k);
        v2f a = { (ae.x + a1.x + a2.x) * inv3,
                  (ae.y + a1.y + a2.y) * inv3 };

        float2 be = *(const float2*)(pe + k);
        float2 b1 = *(const float2*)(p1 + k);
        float2 b2 = *(const float2*)(p2 + k);
        v2f bp = { (be.x + b1.x + b2.x) * inv3,
                   (be.y + b1.y + b2.y) * inv3 };

        float2 ce = *(const float2*)(ge + k);
        float2 c1 = *(const float2*)(g1 + k);
        float2 c2 = *(const float2*)(g2 + k);
        v2f bn = { (ce.x + c1.x + c2.x) * inv3,
                   (ce.y + c1.y + c2.y) * inv3 };

        cp = __builtin_amdgcn_wmma_f32_16x16x4_f32(false, a, false, bp,
                                                   (short)0, cp, false, false);
        cn = __builtin_amdgcn_wmma_f32_16x16x4_f32(false, a, false, bn,
                                                   (short)0, cn, false, false);
    }

    // Diagonal write-out: lanes 0-7 hold D[l,l] in component l;
    // lanes 24-31 hold D[l-16, l-16] in component l-24.
    if (lane < 8) {
        out[wave * 16 + lane]         = lgcn_vext(cp, lane);
        out[BATCH + wave * 16 + lane] = lgcn_vext(cn, lane);
    } else if (lane >= 24) {
        int mm = lane - 16;
        int r  = lane - 24;
        out[wave * 16 + mm]         = lgcn_vext(cp, r);
        out[BATCH + wave * 16 + mm] = lgcn_vext(cn, r);
    }
}

// ---------------------------------------------------------------------------
// Launch: zero ws -> spmm layer1 -> spmm layer2 -> wmma scoring.
// d_ws layout: buf1 = h1 (NUM_NODES*64 f32), buf2 = h2 (NUM_NODES*64 f32).
// ---------------------------------------------------------------------------
extern "C" void kernel_launch(void* const* d_in, const int* in_sizes, int n_in,
                              void* d_out, int out_size, void* d_ws, size_t ws_size,
                              hipStream_t stream) {
    const int*   user = (const int*)d_in[0];
    const int*   pos  = (const int*)d_in[1];
    const int*   neg  = (const int*)d_in[2];
    const int*   erow = (const int*)d_in[3];
    const int*   ecol = (const int*)d_in[4];
    const float* eval = (const float*)d_in[5];
    const float* uemb = (const float*)d_in[6];
    const float* iemb = (const float*)d_in[7];
    float*       out  = (float*)d_out;

    float* buf1 = (float*)d_ws;
    float* buf2 = buf1 + (size_t)NUM_NODES * EMB_DIM;

    // Zero both accumulation buffers (25.6M floats).
    int n4 = (2 * NUM_NODES * EMB_DIM) / 4;
    lgcn_zero_kernel<<<(n4 + 255) / 256, 256, 0, stream>>>((float4*)d_ws, n4);

    // Layer 1: h1 = A @ concat(user_emb, item_emb)
    lgcn_spmm_kernel<<<NUM_EDGES / 8, 256, 0, stream>>>(
        erow, ecol, eval, uemb, iemb, buf1);

    // Layer 2: h2 = A @ h1  (contiguous source, split at NUM_USERS rows)
    lgcn_spmm_kernel<<<NUM_EDGES / 8, 256, 0, stream>>>(
        erow, ecol, eval, buf1, buf1 + (size_t)NUM_USERS * EMB_DIM, buf2);

    // Scoring: 256 waves, 8 waves/block -> 32 blocks. Exact fit, EXEC full.
    lgcn_score_kernel<<<(BATCH / 16) / 8, 256, 0, stream>>>(
        user, pos, neg, uemb, iemb, buf1, buf2, out);
}